// GCNModel_62680752717982
// MI455X (gfx1250) — compile-verified
//
#include <hip/hip_runtime.h>

#define N_NODES 100000
#define N_EDGES 1200000
#define D       64
#define N_LAYERS 3

// pair-row pitch for swizzled weight tiles in LDS:
// 160 mod 64 = 32 banks -> even pair-rows occupy banks 0..31, odd rows 32..63,
// so the two half-waves of a WMMA B-fragment load hit disjoint banks.
#define WPITCH  160

typedef __attribute__((ext_vector_type(2))) float v2f;
typedef __attribute__((ext_vector_type(8))) float v8f;

// ---------------------------------------------------------------------------
// Edge scatter: agg[row[e]] += h[col[e]] * w[e].
// 16 threads per edge, each owning one float4 chunk of the 64-float feature.
// Native fp32 atomics (global_atomic_add_f32) — memory/atomic bound phase.
// ---------------------------------------------------------------------------
__global__ __launch_bounds__(256)
void sage_scatter(const float* __restrict__ h,
                  const int*   __restrict__ rowi,
                  const int*   __restrict__ coli,
                  const float* __restrict__ ew,
                  float*       __restrict__ agg)
{
    int t = blockIdx.x * blockDim.x + threadIdx.x;
    int e = t >> 4;
    if (e >= N_EDGES) return;
    int c = (t & 15) << 2;

    int   src = coli[e];
    int   dst = rowi[e];
    float w   = ew[e];

    const float4 v = *(const float4*)(h + (size_t)src * D + c);
    float* o = agg + (size_t)dst * D + c;
    unsafeAtomicAdd(o + 0, v.x * w);
    unsafeAtomicAdd(o + 1, v.y * w);
    unsafeAtomicAdd(o + 2, v.z * w);
    unsafeAtomicAdd(o + 3, v.w * w);
}

// ---------------------------------------------------------------------------
// Fused dual-GEMM + bias + ReLU using V_WMMA_F32_16X16X4_F32 (fp32 exact):
//   out = relu(h @ Wself + agg @ Wneigh + b)
// 256 threads = 8 waves; each wave owns a 16-row x 64-col output stripe.
// Weights are staged in LDS pre-swizzled into K-pair-major layout so each
// B-fragment {W[kk][nc], W[kk+1][nc]} is ONE aligned ds_load_b64:
//   sW[(k>>1)*WPITCH + nc*2 + (k&1)]
// A-fragments are aligned 8B float2 global loads matching the 16x4 fp32
// A-operand layout (VGPR0/1 = K=2g / 2g+1, lanes 0-15 = rows M=0..15).
// ---------------------------------------------------------------------------
__global__ __launch_bounds__(256)
void sage_gemm(const float* __restrict__ h,
               const float* __restrict__ agg,
               const float* __restrict__ Wself,
               const float* __restrict__ Wneigh,
               const float* __restrict__ bias,
               float*       __restrict__ out)
{
    __shared__ __align__(16) float sWs[32 * WPITCH];   // 20 KB, swizzled
    __shared__ __align__(16) float sWn[32 * WPITCH];   // 20 KB, swizzled
    __shared__ float sB[D];

    const int tid       = threadIdx.x;
    const int block_row = blockIdx.x * 128;

    // cooperative weight load with swizzle: global float4 -> 4 scalar LDS
    // stores at sW[(k>>1)*WPITCH + n*2 + (k&1)]
    for (int i = tid * 4; i < D * D; i += 256 * 4) {
        float4 ws = *(const float4*)&Wself[i];
        float4 wn = *(const float4*)&Wneigh[i];
        int k = i >> 6;
        int n = i & 63;
        int base = (k >> 1) * WPITCH + (k & 1);
        sWs[base + (n + 0) * 2] = ws.x;
        sWs[base + (n + 1) * 2] = ws.y;
        sWs[base + (n + 2) * 2] = ws.z;
        sWs[base + (n + 3) * 2] = ws.w;
        sWn[base + (n + 0) * 2] = wn.x;
        sWn[base + (n + 1) * 2] = wn.y;
        sWn[base + (n + 2) * 2] = wn.z;
        sWn[base + (n + 3) * 2] = wn.w;
    }
    if (tid < D) sB[tid] = bias[tid];
    __syncthreads();

    const int wave = tid >> 5;         // 0..7 -> 16-row stripe
    const int lane = tid & 31;
    const int g    = lane >> 4;        // half-wave group
    const int lr   = lane & 15;

    int arow = block_row + wave * 16 + lr;          // A-fragment source row
    if (arow >= N_NODES) arow = N_NODES - 1;        // clamp (stores are guarded)
    const float* hrow   = h   + (size_t)arow * D;
    const float* aggrow = agg + (size_t)arow * D;

    // accumulators: 4 N-tiles of 16x16; init with bias (col = n*16+lr for all 8 regs)
    v8f acc[4];
#pragma unroll
    for (int n = 0; n < 4; ++n) {
        float bv = sB[n * 16 + lr];
#pragma unroll
        for (int j = 0; j < 8; ++j) acc[n][j] = bv;
    }

#pragma unroll
    for (int k = 0; k < D; k += 4) {
        const int kk = k + 2 * g;                    // this half-wave's K base
        const int p  = kk >> 1;                      // pair-row in swizzled LDS
        v2f aS = { hrow[kk],   hrow[kk + 1]   };     // one global_load_b64
        v2f aN = { aggrow[kk], aggrow[kk + 1] };     // one global_load_b64
#pragma unroll
        for (int n = 0; n < 4; ++n) {
            const int nc = n * 16 + lr;
            v2f bS = *(const v2f*)&sWs[p * WPITCH + nc * 2];   // one ds_load_b64
            v2f bN = *(const v2f*)&sWn[p * WPITCH + nc * 2];   // one ds_load_b64
            acc[n] = __builtin_amdgcn_wmma_f32_16x16x4_f32(
                false, aS, false, bS, (short)0, acc[n], false, false);
            acc[n] = __builtin_amdgcn_wmma_f32_16x16x4_f32(
                false, aN, false, bN, (short)0, acc[n], false, false);
        }
    }

    // ReLU + store; C/D layout: VGPR j -> row m_base + j (m_base includes 8*g), col n*16+lr
    const int m_base = block_row + wave * 16 + 8 * g;
#pragma unroll
    for (int n = 0; n < 4; ++n) {
#pragma unroll
        for (int j = 0; j < 8; ++j) {
            int grow = m_base + j;
            if (grow < N_NODES) {
                float v = acc[n][j];
                out[(size_t)grow * D + n * 16 + lr] = v > 0.0f ? v : 0.0f;
            }
        }
    }
}

// ---------------------------------------------------------------------------
// Driver: 3 layers of { zero agg, scatter, fused WMMA gemm }.
// ws layout: [agg: N*D][hA: N*D]; d_out doubles as ping-pong buffer:
//   layer0: x  -> out ; layer1: out -> hA ; layer2: hA -> out
// ---------------------------------------------------------------------------
extern "C" void kernel_launch(void* const* d_in, const int* in_sizes, int n_in,
                              void* d_out, int out_size, void* d_ws, size_t ws_size,
                              hipStream_t stream)
{
    const float* x       = (const float*)d_in[0];
    const int*   eidx    = (const int*)  d_in[1];   // [2, E]
    const float* ew      = (const float*)d_in[2];
    const float* self_k  = (const float*)d_in[3];   // [3, 64, 64]
    const float* neigh_k = (const float*)d_in[4];   // [3, 64, 64]
    const float* biases  = (const float*)d_in[5];   // [3, 64]
    float*       out     = (float*)d_out;

    const int* rowi = eidx;
    const int* coli = eidx + N_EDGES;

    float* agg = (float*)d_ws;
    float* hA  = agg + (size_t)N_NODES * D;

    const size_t agg_bytes = (size_t)N_NODES * D * sizeof(float);

    const float* hin = x;
    float* houts[N_LAYERS] = { out, hA, out };

    dim3 sblk(256), gblk(256);
    dim3 sgrid((N_EDGES * 16 + 255) / 256);
    dim3 ggrid((N_NODES + 127) / 128);

    for (int l = 0; l < N_LAYERS; ++l) {
        hipMemsetAsync(agg, 0, agg_bytes, stream);
        sage_scatter<<<sgrid, sblk, 0, stream>>>(hin, rowi, coli, ew, agg);
        sage_gemm<<<ggrid, gblk, 0, stream>>>(hin, agg,
                                              self_k + l * D * D,
                                              neigh_k + l * D * D,
                                              biases + l * D,
                                              houts[l]);
        hin = houts[l];
    }
}